// UncertaintyGuidedCorrector_35527969473145
// MI455X (gfx1250) — compile-verified
//
#include <hip/hip_runtime.h>

// Problem constants (from reference)
#define B_   8
#define T_   316
#define CIN_ 64
#define HF_  128
#define G_   32
#define E_   384
#define OUT_ 8

typedef float v2f __attribute__((ext_vector_type(2)));
typedef float v8f __attribute__((ext_vector_type(8)));

// Bilinear upsample weight matrix W[i, b] for pf in {4, 8}; W = I for pf = 8.
// Matches reference: rel = clip((i+0.5)*pf/8 - 0.5, 0, pf-1), i0 = floor(rel),
// w = rel - i0, i1 = min(i0+1, pf-1); row i has (1-w) at i0 and w at i1.
__device__ __forceinline__ float wmat(int pf, int i, int b) {
    if (i >= 8 || b >= pf || b < 0) return 0.0f;
    float pff  = (float)pf;
    float rel  = ((float)i + 0.5f) * (pff * 0.125f) - 0.5f;
    rel        = fminf(fmaxf(rel, 0.0f), pff - 1.0f);
    float i0f  = floorf(rel);
    float w    = rel - i0f;
    float i1f  = fminf(i0f + 1.0f, pff - 1.0f);
    int   i0   = (int)i0f;
    int   i1   = (int)i1f;
    float acc  = 0.0f;
    if (b == i0) acc += 1.0f - w;
    if (b == i1) acc += w;
    return acc;
}

// ---------------------------------------------------------------------------
// Kernel 1: cx, cy, scale_idx
// ---------------------------------------------------------------------------
__global__ void k_meta(const int* __restrict__ metas,
                       float* __restrict__ cx, float* __restrict__ cy,
                       float* __restrict__ sc) {
    int t = blockIdx.x * blockDim.x + threadIdx.x;
    if (t >= B_ * T_) return;
    const int* m = metas + (size_t)t * 5;
    float half = (float)m[2] * 0.5f;
    cx[t] = (float)m[1] + half;
    cy[t] = (float)m[0] + half;
    int p = m[3];
    sc[t] = (p == 16) ? 1.0f : ((p == 8) ? 2.0f : 0.0f);
}

// ---------------------------------------------------------------------------
// Kernel 2a/2b: owner map init + scatter (cells are disjoint by construction)
// ---------------------------------------------------------------------------
__global__ void k_owner_init(int* __restrict__ owner) {
    int i = blockIdx.x * blockDim.x + threadIdx.x;
    if (i < B_ * G_ * G_) owner[i] = -1;
}

__global__ void k_scatter(const int* __restrict__ metas, int* __restrict__ owner) {
    int bt = blockIdx.x * blockDim.x + threadIdx.x;
    if (bt >= B_ * T_) return;
    int b = bt / T_, t = bt % T_;
    const int* m = metas + (size_t)bt * 5;
    int r = m[0], c = m[1], span = m[2], p = m[3];
    if (p <= 0) return;
    for (int di = 0; di < span && di < 4; ++di)
        for (int dj = 0; dj < span && dj < 4; ++dj) {
            int rr = r + di, cc = c + dj;
            if (rr < G_ && cc < G_)
                owner[b * G_ * G_ + rr * G_ + cc] = t;
        }
}

// ---------------------------------------------------------------------------
// Kernel 3: fmap gather  (b, e, gr, gc) row-major; tid fastest dim = gc
// ---------------------------------------------------------------------------
__global__ void k_fmap(const float* __restrict__ tokens,
                       const int* __restrict__ owner,
                       float* __restrict__ fmap) {
    long long idx = (long long)blockIdx.x * blockDim.x + threadIdx.x;
    const long long total = (long long)B_ * E_ * G_ * G_;
    if (idx >= total) return;
    int gc = (int)(idx & (G_ - 1));
    int gr = (int)((idx >> 5) & (G_ - 1));
    int e  = (int)((idx / (G_ * G_)) % E_);
    int b  = (int)(idx / ((long long)E_ * G_ * G_));
    int o  = owner[b * G_ * G_ + gr * G_ + gc];
    float v = (o >= 0) ? tokens[((size_t)b * T_ + o) * E_ + e] : 0.0f;
    fmap[idx] = v;
}

// ---------------------------------------------------------------------------
// Kernel 4: patch extraction via WMMA.
//   patch = W * win * W^T as two chained 16x16x4 f32 WMMA pairs.
//   One block (8 wave32) per token; each wave handles 8 channels.
//   Window loads are branch-free: clamped 8B-aligned b64 loads * 0/1 mask.
// ---------------------------------------------------------------------------
__global__ void __launch_bounds__(256)
k_patch(const float* __restrict__ x_in, const int* __restrict__ metas,
        float* __restrict__ patches) {
    int bt   = blockIdx.x;            // 0 .. B*T-1
    int b    = bt / T_;
    int wave = threadIdx.x >> 5;      // 0..7
    int l    = threadIdx.x & 31;      // lane
    int m    = l & 15;                // A/B: M row or N col
    int kh   = l >> 4;                // lane half -> K sub-chunk

    const int* mp = metas + (size_t)bt * 5;
    int r = mp[0], c = mp[1], p = mp[3];
    int pf = p >> 1;                  // p_feat = p // DOWN
    if (pf < 1) pf = 1;               // maximum(p_feat, 1)
    float gate = (p > 0) ? 1.0f : 0.0f;   // where(p_px > 0, patch, 0)
    int off_r = r * 4, off_c = c * 4;     // STRIDE = 4 (both even)

    // Weight operand, dual use:
    //   step A B-operand (W^T, lane = N col) and step B A-operand (W, lane = M row)
    //   both reduce per-lane to W[m, k].
    v2f wlo, whi;
    wlo.x = wmat(pf, m, 2 * kh + 0) * gate;
    wlo.y = wmat(pf, m, 2 * kh + 1) * gate;
    whi.x = wmat(pf, m, 4 + 2 * kh) * gate;
    whi.y = wmat(pf, m, 5 + 2 * kh) * gate;

    // Loop-invariant per-lane window addressing (always in-bounds after clamp;
    // clamped positions carry mask 0).  Column pair bases are even -> 8B aligned.
    int y   = min(off_r + m, HF_ - 1);
    int x0  = min(off_c + 2 * kh,     HF_ - 2);
    int x1  = min(off_c + 4 + 2 * kh, HF_ - 2);
    int o0  = y * HF_ + x0;           // chunk0 pair offset within channel plane
    int o1  = y * HF_ + x1;           // chunk1 pair offset within channel plane
    // 0/1 masks: win[m, b] is live iff m < pf && b < pf.
    float rowm = (m < pf) ? 1.0f : 0.0f;
    v2f mk0, mk1;
    mk0.x = rowm * ((2 * kh + 0 < pf) ? 1.0f : 0.0f);
    mk0.y = rowm * ((2 * kh + 1 < pf) ? 1.0f : 0.0f);
    mk1.x = rowm * ((4 + 2 * kh < pf) ? 1.0f : 0.0f);
    mk1.y = rowm * ((5 + 2 * kh < pf) ? 1.0f : 0.0f);

    const float* img  = x_in    + (size_t)b  * CIN_ * HF_ * HF_;
    float*       outp = patches + (size_t)bt * CIN_ * OUT_ * OUT_;

    const int c0 = wave * 8;
    for (int cc = c0; cc < c0 + 8; ++cc) {
        const float* ch = img + (size_t)cc * HF_ * HF_;

        // Prefetch next channel's window row (clamped channel -> in-bounds).
        const float* chn = img + (size_t)min(cc + 1, CIN_ - 1) * HF_ * HF_;
        __builtin_prefetch(chn + o0, 0, 1);

        // Step A: U = win * W^T.  A[m, k] = win[m, k], branch-free masked loads.
        v2f a0 = *(const v2f*)(ch + o0);
        v2f a1 = *(const v2f*)(ch + o1);
        a0.x *= mk0.x;  a0.y *= mk0.y;
        a1.x *= mk1.x;  a1.y *= mk1.y;

        v8f u = {0.f, 0.f, 0.f, 0.f, 0.f, 0.f, 0.f, 0.f};
        u = __builtin_amdgcn_wmma_f32_16x16x4_f32(false, a0, false, wlo,
                                                  (short)0, u, false, false);
        u = __builtin_amdgcn_wmma_f32_16x16x4_f32(false, a1, false, whi,
                                                  (short)0, u, false, false);

        // Re-layout U (D: lane = j, vgpr = a) into step-B B-operand
        // (lane = j, vgpr/half = a). All lanes execute every shfl.
        float s0 = __shfl(u[0], m, 32);
        float s1 = __shfl(u[1], m, 32);
        float s2 = __shfl(u[2], m, 32);
        float s3 = __shfl(u[3], m, 32);
        float s4 = __shfl(u[4], m, 32);
        float s5 = __shfl(u[5], m, 32);
        float s6 = __shfl(u[6], m, 32);
        float s7 = __shfl(u[7], m, 32);
        v2f b0, b1;
        b0.x = kh ? s2 : s0;  b0.y = kh ? s3 : s1;
        b1.x = kh ? s6 : s4;  b1.y = kh ? s7 : s5;

        // Step B: patch = W * U
        v8f d = {0.f, 0.f, 0.f, 0.f, 0.f, 0.f, 0.f, 0.f};
        d = __builtin_amdgcn_wmma_f32_16x16x4_f32(false, wlo, false, b0,
                                                  (short)0, d, false, false);
        d = __builtin_amdgcn_wmma_f32_16x16x4_f32(false, whi, false, b1,
                                                  (short)0, d, false, false);

        // D: patch[i, j] at vgpr i, lane j (lanes 0-7 valid columns).
        if (l < 8) {
            float* o = outp + (size_t)cc * OUT_ * OUT_ + l;
            o[0]  = d[0]; o[8]  = d[1]; o[16] = d[2]; o[24] = d[3];
            o[32] = d[4]; o[40] = d[5]; o[48] = d[6]; o[56] = d[7];
        }
    }
}

// ---------------------------------------------------------------------------
extern "C" void kernel_launch(void* const* d_in, const int* in_sizes, int n_in,
                              void* d_out, int out_size, void* d_ws, size_t ws_size,
                              hipStream_t stream) {
    const float* x_in   = (const float*)d_in[0];   // (8, 64, 128, 128)
    const float* tokens = (const float*)d_in[1];   // (8, 316, 384)
    const int*   metas  = (const int*)d_in[2];     // (8, 316, 5)

    float* out = (float*)d_out;
    const size_t P1 = (size_t)B_ * T_ * CIN_ * OUT_ * OUT_;   // patches
    float* patches = out;
    float* cx      = out + P1;
    float* cy      = cx + (size_t)B_ * T_;
    float* sc      = cy + (size_t)B_ * T_;
    float* fmap    = sc + (size_t)B_ * T_;

    int* owner = (int*)d_ws;   // B*G*G ints (32 KB)

    k_meta<<<(B_ * T_ + 255) / 256, 256, 0, stream>>>(metas, cx, cy, sc);
    k_owner_init<<<(B_ * G_ * G_ + 255) / 256, 256, 0, stream>>>(owner);
    k_scatter<<<(B_ * T_ + 255) / 256, 256, 0, stream>>>(metas, owner);
    k_fmap<<<(int)(((long long)B_ * E_ * G_ * G_ + 255) / 256), 256, 0, stream>>>(
        tokens, owner, fmap);
    k_patch<<<B_ * T_, 256, 0, stream>>>(x_in, metas, patches);
}